// KnowledgeGroundedModel_15333033247402
// MI455X (gfx1250) — compile-verified
//
#include <hip/hip_runtime.h>
#include <math.h>

// Problem constants (match reference)
#define LSEQ 32
#define BB   128
#define FF   4096
#define EE   512
#define HH   512
#define NLL  2
#define KK   20

typedef __attribute__((ext_vector_type(2))) float v2f;
typedef __attribute__((ext_vector_type(8))) float v8f;

__device__ __forceinline__ float waveReduceSum(float v) {
#pragma unroll
  for (int m = 16; m >= 1; m >>= 1) v += __shfl_xor(v, m, 32);
  return v;
}

// ---------------------------------------------------------------------------
// 1) q[b,e] = mean_l enc_emb[ids[l,b], e]
// ---------------------------------------------------------------------------
__global__ void k_query(const int* __restrict__ ids, const float* __restrict__ emb,
                        float* __restrict__ q) {
  int i = blockIdx.x * blockDim.x + threadIdx.x;   // over B*E
  if (i >= BB * EE) return;
  int b = i / EE, e = i - b * EE;
  float s = 0.f;
#pragma unroll
  for (int l = 0; l < LSEQ; ++l) {
    int tok = ids[l * BB + b];
    s += emb[(size_t)tok * EE + e];
  }
  q[i] = s * (1.0f / LSEQ);
}

// ---------------------------------------------------------------------------
// 2) scores[b,f] = dot(q[b,:], facts[b,f,:])   -- streams 1 GB, HBM-bound
//    one wave handles 8 fact rows; lanes load float4, wave-reduce.
// ---------------------------------------------------------------------------
__global__ void k_scores(const float* __restrict__ q, const float* __restrict__ facts,
                         float* __restrict__ scores) {
  __shared__ float qs[EE];
  int b = blockIdx.y;
  for (int e = threadIdx.x; e < EE; e += blockDim.x) qs[e] = q[b * EE + e];
  __syncthreads();

  int wave = threadIdx.x >> 5;
  int lane = threadIdx.x & 31;
  int f0 = blockIdx.x * 64 + wave * 8;
  const float* base = facts + (size_t)b * FF * EE;

  for (int fi = 0; fi < 8; ++fi) {
    int f = f0 + fi;
    const float* row = base + (size_t)f * EE;
    // prefetch next row while computing this one
    __builtin_prefetch(row + EE + lane * 4, 0, 0);
    float acc = 0.f;
#pragma unroll
    for (int it = 0; it < 4; ++it) {
      int e = it * 128 + lane * 4;
      float4 fv = *(const float4*)(row + e);
      acc += fv.x * qs[e] + fv.y * qs[e + 1] + fv.z * qs[e + 2] + fv.w * qs[e + 3];
    }
    acc = waveReduceSum(acc);
    if (lane == 0) scores[(size_t)b * FF + f] = acc;
  }
}

// ---------------------------------------------------------------------------
// 3) top-k (K=20) per batch row: iterative block-wide argmax in LDS.
//    Tie-break lower index, values emitted descending (matches jax.lax.top_k).
// ---------------------------------------------------------------------------
__global__ void k_topk(const float* __restrict__ scores, int* __restrict__ idx) {
  __shared__ float vals[FF];
  __shared__ float rv[256];
  __shared__ int   ri[256];
  int b = blockIdx.x;
  for (int f = threadIdx.x; f < FF; f += 256) vals[f] = scores[(size_t)b * FF + f];
  __syncthreads();

  for (int t = 0; t < KK; ++t) {
    float best = -INFINITY; int bi = 0x7fffffff;
    for (int f = threadIdx.x; f < FF; f += 256) {
      float v = vals[f];
      if (v > best || (v == best && f < bi)) { best = v; bi = f; }
    }
    rv[threadIdx.x] = best; ri[threadIdx.x] = bi;
    __syncthreads();
    for (int s = 128; s >= 1; s >>= 1) {
      if (threadIdx.x < (unsigned)s) {
        float ov = rv[threadIdx.x + s]; int oi = ri[threadIdx.x + s];
        if (ov > rv[threadIdx.x] || (ov == rv[threadIdx.x] && oi < ri[threadIdx.x])) {
          rv[threadIdx.x] = ov; ri[threadIdx.x] = oi;
        }
      }
      __syncthreads();
    }
    if (threadIdx.x == 0) {
      idx[b * KK + t] = ri[0];
      vals[ri[0]] = -INFINITY;
    }
    __syncthreads();
  }
}

// ---------------------------------------------------------------------------
// 4) topf[b,k,:] = facts[b, idx[b,k], :]
// ---------------------------------------------------------------------------
__global__ void k_gather(const float* __restrict__ facts, const int* __restrict__ idx,
                         float* __restrict__ topf) {
  int i = blockIdx.x * blockDim.x + threadIdx.x;   // over B*K*E
  if (i >= BB * KK * EE) return;
  int e = i & (EE - 1);
  int bk = i / EE;
  int b = bk / KK;
  int id = idx[bk];
  topf[i] = facts[((size_t)b * FF + id) * EE + e];
}

// ---------------------------------------------------------------------------
// 5) fact_M / fact_C = topf[2560,512] x W^T[512,512] + bias, via
//    V_WMMA_F32_16X16X4_F32. Each wave: one 16x64 output strip, 4 accumulators.
//    A frag (16x4):  lane -> M = lane&15, K = 2*(lane>>4) + {0,1}
//    B frag (4x16):  lane -> N = lane&15, K = 2*(lane>>4) + {0,1}; B[e,n]=W[n,e]
//    C/D: vgpr r -> row (r + 8*(lane>>4)), col lane&15
// ---------------------------------------------------------------------------
__global__ void k_fact_gemm(const float* __restrict__ A,
                            const float* __restrict__ Aw, const float* __restrict__ Ab,
                            const float* __restrict__ Cw, const float* __restrict__ Cb,
                            float* __restrict__ outM, float* __restrict__ outC) {
  const float* W  = blockIdx.y ? Cw : Aw;
  const float* Bi = blockIdx.y ? Cb : Ab;
  float* Out      = blockIdx.y ? outC : outM;

  int wid = blockIdx.x * (blockDim.x >> 5) + (threadIdx.x >> 5);  // 0..1279
  int mtile  = wid >> 3;   // 0..159
  int nstrip = wid & 7;    // 0..7
  int m0 = mtile * 16;
  int n0 = nstrip * 64;
  int lane = threadIdx.x & 31;
  int lo = lane & 15, hi = lane >> 4;

  const float* aRow  = A + (size_t)(m0 + lo) * EE + 2 * hi;
  const float* wRow0 = W + (size_t)(n0 + 0 * 16 + lo) * EE + 2 * hi;
  const float* wRow1 = W + (size_t)(n0 + 1 * 16 + lo) * EE + 2 * hi;
  const float* wRow2 = W + (size_t)(n0 + 2 * 16 + lo) * EE + 2 * hi;
  const float* wRow3 = W + (size_t)(n0 + 3 * 16 + lo) * EE + 2 * hi;

  v8f acc0 = {}, acc1 = {}, acc2 = {}, acc3 = {};
  for (int k = 0; k < EE; k += 4) {
    v2f a  = *(const v2f*)(aRow  + k);
    v2f b0 = *(const v2f*)(wRow0 + k);
    v2f b1 = *(const v2f*)(wRow1 + k);
    v2f b2 = *(const v2f*)(wRow2 + k);
    v2f b3 = *(const v2f*)(wRow3 + k);
    acc0 = __builtin_amdgcn_wmma_f32_16x16x4_f32(false, a, false, b0, (short)0, acc0, false, false);
    acc1 = __builtin_amdgcn_wmma_f32_16x16x4_f32(false, a, false, b1, (short)0, acc1, false, false);
    acc2 = __builtin_amdgcn_wmma_f32_16x16x4_f32(false, a, false, b2, (short)0, acc2, false, false);
    acc3 = __builtin_amdgcn_wmma_f32_16x16x4_f32(false, a, false, b3, (short)0, acc3, false, false);
  }

  int c0 = n0 + lo;
  float bia0 = Bi[c0], bia1 = Bi[c0 + 16], bia2 = Bi[c0 + 32], bia3 = Bi[c0 + 48];
#pragma unroll
  for (int r = 0; r < 8; ++r) {
    size_t row = (size_t)(m0 + r + 8 * hi) * HH;
    Out[row + c0]      = acc0[r] + bia0;
    Out[row + c0 + 16] = acc1[r] + bia1;
    Out[row + c0 + 32] = acc2[r] + bia2;
    Out[row + c0 + 48] = acc3[r] + bia3;
  }
}

// ---------------------------------------------------------------------------
// 6) update(u): P = softmax_k(topf[b,k,:] . u[l,b,:]); o = P @ topf; out = o+u
//    one wave per (tensor, l, b) row; 16 elems per lane, stride-32 coalesced.
// ---------------------------------------------------------------------------
__global__ void k_update(const float* __restrict__ h, const float* __restrict__ c,
                         const float* __restrict__ topf, float* __restrict__ out) {
  int wid  = blockIdx.x * (blockDim.x >> 5) + (threadIdx.x >> 5);  // 0..511
  int lane = threadIdx.x & 31;
  int t   = wid >> 8;       // 0: h, 1: c
  int rem = wid & 255;      // l*B + b
  int b   = rem & (BB - 1);

  const float* u = (t ? c : h) + (size_t)rem * HH;
  float* o = out + (size_t)t * (NLL * BB * HH) + (size_t)rem * HH;

  float ur[16];
#pragma unroll
  for (int j = 0; j < 16; ++j) ur[j] = u[lane + j * 32];

  const float* tf = topf + (size_t)b * KK * EE;
  float s[KK];
  for (int k = 0; k < KK; ++k) {
    const float* row = tf + (size_t)k * EE;
    float p = 0.f;
#pragma unroll
    for (int j = 0; j < 16; ++j) p += row[lane + j * 32] * ur[j];
    s[k] = waveReduceSum(p);
  }
  float m = s[0];
  for (int k = 1; k < KK; ++k) m = fmaxf(m, s[k]);
  float den = 0.f;
  for (int k = 0; k < KK; ++k) { s[k] = expf(s[k] - m); den += s[k]; }
  float inv = 1.f / den;

  float acc[16];
#pragma unroll
  for (int j = 0; j < 16; ++j) acc[j] = 0.f;
  for (int k = 0; k < KK; ++k) {
    const float* row = tf + (size_t)k * EE;
    float p = s[k] * inv;
#pragma unroll
    for (int j = 0; j < 16; ++j) acc[j] += p * row[lane + j * 32];
  }
#pragma unroll
  for (int j = 0; j < 16; ++j) o[lane + j * 32] = ur[j] + acc[j];
}

// ---------------------------------------------------------------------------
extern "C" void kernel_launch(void* const* d_in, const int* in_sizes, int n_in,
                              void* d_out, int out_size, void* d_ws, size_t ws_size,
                              hipStream_t stream) {
  const int*   ids   = (const int*)d_in[0];
  const float* facts = (const float*)d_in[1];
  const float* h     = (const float*)d_in[2];
  const float* c     = (const float*)d_in[3];
  const float* emb   = (const float*)d_in[4];
  const float* Aw    = (const float*)d_in[5];
  const float* Ab    = (const float*)d_in[6];
  const float* Cw    = (const float*)d_in[7];
  const float* Cb    = (const float*)d_in[8];
  float* out = (float*)d_out;

  // workspace layout (floats): q[65536] | scores[524288] | idx[2560 ints] | topf[1310720]
  float* ws     = (float*)d_ws;
  float* q      = ws;
  float* scores = ws + BB * EE;
  int*   idx    = (int*)(scores + (size_t)BB * FF);
  float* topf   = (float*)(idx + BB * KK);

  // output layout: update(h)[262144? no: 131072] | update(c)[131072] | fact_M[1310720] | fact_C[1310720]
  float* outM = out + 2 * NLL * BB * HH;
  float* outC = outM + BB * KK * HH;

  k_query <<<dim3((BB * EE) / 256), 256, 0, stream>>>(ids, emb, q);
  k_scores<<<dim3(FF / 64, BB),     256, 0, stream>>>(q, facts, scores);
  k_topk  <<<dim3(BB),              256, 0, stream>>>(scores, idx);
  k_gather<<<dim3((BB * KK * EE) / 256), 256, 0, stream>>>(facts, idx, topf);
  k_fact_gemm<<<dim3(320, 2), 128, 0, stream>>>(topf, Aw, Ab, Cw, Cb, outM, outC);
  k_update<<<dim3(64), 256, 0, stream>>>(h, c, topf, out);
}